// MeshPool_6468220748707
// MI455X (gfx1250) — compile-verified
//
#include <hip/hip_runtime.h>
#include <stdint.h>

#define F_DIM 512

typedef float v2f __attribute__((ext_vector_type(2)));
typedef float v8f __attribute__((ext_vector_type(8)));

// ---------------------------------------------------------------------------
// 1) Row squared-norms via V_WMMA_F32_16X16X4_F32.
//    One wave handles 16 rows. A (16x4 f32) layout: lanes 0-15 = M, VGPR0/1 =
//    K0/K1; lanes 16-31 = M, K2/K3.  B = A^T has the identical register image
//    (lane = N), so we pass the same v2f for both operands and accumulate the
//    16x16 Gram; its diagonal is sq[] in exact fp32 (ordering-critical).
// ---------------------------------------------------------------------------
__global__ __launch_bounds__(128) void sq_kernel(const float* __restrict__ img,
                                                 float* __restrict__ sq, int V) {
  __shared__ float tile[4][32][8];
  const int lane = threadIdx.x & 31;
  const int wave = threadIdx.x >> 5;
  const int vbase = (blockIdx.x * 4 + wave) * 16;
  const bool active = (vbase < V);

  v8f acc = {0.f, 0.f, 0.f, 0.f, 0.f, 0.f, 0.f, 0.f};
  if (active) {
    const int row = vbase + (lane & 15);
    const int koff = (lane >> 4) * 2;  // lanes 16-31 hold K=2,3
    const float* p = img + (size_t)row * F_DIM + koff;
    for (int k = 0; k < F_DIM; k += 4) {
      v2f a = *(const v2f*)(p + k);
#if __has_builtin(__builtin_amdgcn_wmma_f32_16x16x4_f32)
      acc = __builtin_amdgcn_wmma_f32_16x16x4_f32(
          /*neg_a=*/false, a, /*neg_b=*/false, a,
          /*c_mod=*/(short)0, acc, /*reuse_a=*/false, /*reuse_b=*/false);
#else
      acc[0] += a.x * a.x + a.y * a.y;  // fallback (never expected on gfx1250)
#endif
    }
#pragma unroll
    for (int r = 0; r < 8; ++r) tile[wave][lane][r] = acc[r];
  }
  __syncthreads();
  if (active && lane < 16) {
    // D layout: VGPR r, lanes 0-15 -> (M=r,N=lane); lanes 16-31 -> (M=r+8,N=lane-16)
    float d = (lane < 8) ? tile[wave][lane][lane] : tile[wave][lane + 16][lane - 8];
    sq[vbase + lane] = d;
  }
}

// ---------------------------------------------------------------------------
// 2) Init: zero adjacency bitset, mask=1, boundary flags from vs, collapsed=0
// ---------------------------------------------------------------------------
__global__ void init_kernel(unsigned* U, size_t uwords, int* mask, int* bnd,
                            int* collapsed, const float* __restrict__ vs,
                            int V, int NSORT) {
  size_t i = (size_t)blockIdx.x * blockDim.x + threadIdx.x;
  if (i < uwords) U[i] = 0u;
  if (i < (size_t)V) {
    mask[i] = 1;
    float x = vs[i * 2 + 0], y = vs[i * 2 + 1];
    bnd[i] = (x < 1e-3f) || (x > 1.0f - 1e-3f) || (y < 1e-3f) || (y > 1.0f - 1e-3f);
  }
  if (i < (size_t)NSORT) collapsed[i] = 0;
}

// Symmetric adjacency U = adj | adj^T (exact under the reference's merge ops)
__global__ void adj_kernel(const int* __restrict__ edges, unsigned* U, int E, int wpr) {
  int i = blockIdx.x * blockDim.x + threadIdx.x;
  if (i >= E) return;
  int v0 = edges[i], v1 = edges[E + i];
  atomicOr(&U[(size_t)v0 * wpr + (v1 >> 5)], 1u << (v1 & 31));
  atomicOr(&U[(size_t)v1 * wpr + (v0 >> 5)], 1u << (v0 & 31));
}

// ---------------------------------------------------------------------------
// 3) Pack stable sort keys: key = monotone_u32(prio) : edge_id.
//    prio >= 0 here but the mapping is fully monotone over IEEE floats.
// ---------------------------------------------------------------------------
__global__ void pack_kernel(const float* __restrict__ sq, const int* __restrict__ edges,
                            unsigned long long* keys, int E, int NSORT) {
  int i = blockIdx.x * blockDim.x + threadIdx.x;
  if (i >= NSORT) return;
  unsigned long long key = ~0ULL;  // padding sorts last
  if (i < E) {
    int v0 = edges[i], v1 = edges[E + i];
    float p = sq[v0] + sq[v1];
    unsigned u = __float_as_uint(p);
    u = (u & 0x80000000u) ? ~u : (u | 0x80000000u);
    key = ((unsigned long long)u << 32) | (unsigned)i;
  }
  keys[i] = key;
}

// ---------------------------------------------------------------------------
// 4) Full bitonic sort of 32768 u64 keys in 256KB LDS — single workgroup.
//    CDNA5 allows up to 320KB LDS per workgroup; zero global traffic between
//    the ~120 compare/exchange passes.
// ---------------------------------------------------------------------------
__global__ __launch_bounds__(1024) void bitonic_kernel(unsigned long long* keys, int N) {
  extern __shared__ unsigned long long sk[];
  const int tid = threadIdx.x;
  for (int i = tid; i < N; i += 1024) sk[i] = keys[i];
  __syncthreads();
  for (int k = 2; k <= N; k <<= 1) {
    for (int j = k >> 1; j > 0; j >>= 1) {
      for (int i = tid; i < N; i += 1024) {
        int ixj = i ^ j;
        if (ixj > i) {
          unsigned long long a = sk[i], b = sk[ixj];
          bool up = ((i & k) == 0);
          if ((a > b) == up) { sk[i] = b; sk[ixj] = a; }
        }
      }
      __syncthreads();
    }
  }
  for (int i = tid; i < N; i += 1024) keys[i] = sk[i];
}

// ---------------------------------------------------------------------------
// 5) Sequential greedy decimation — one wave. Neighbor sets are bitset rows
//    (wpr words, 32 lanes x nw words/lane); shared-neighbor count is an
//    AND+popcount + 5-step shfl_xor wave reduce. The 10.6MB bitset stays hot
//    in the 192MB L2. Symmetric maintenance:
//      U[v0] = (U[v0]|U[v1]) \ {v0,v1};  U[v1]=0;
//      for u in U[v0]: set U[u][v0], clear U[u][v1]   (scatter via atomics)
// ---------------------------------------------------------------------------
__global__ __launch_bounds__(32) void decimate_kernel(
    const unsigned long long* __restrict__ keys, const int* __restrict__ edges,
    unsigned* U, int* mask, const int* __restrict__ bnd, int* collapsed,
    int V, int E, int wpr, int nw, int target) {
  const int lane = threadIdx.x;
  int count = V;
  for (int ptr = 0; ptr < E && count > target; ++ptr) {
    int eid = (int)(unsigned)(keys[ptr] & 0xFFFFFFFFull);
    int v0 = edges[eid], v1 = edges[E + eid];
    if (!(mask[v0] && mask[v1])) continue;
    if (bnd[v0] || bnd[v1]) continue;

    unsigned* r0 = U + (size_t)v0 * wpr;
    unsigned* r1 = U + (size_t)v1 * wpr;
    unsigned w0[16], w1[16];
    int common = 0;
    for (int i = 0; i < nw; ++i) {
      int c = i * 32 + lane;
      w0[i] = r0[c];
      w1[i] = r1[c];
      common += __popc(w0[i] & w1[i]);
    }
    for (int off = 16; off; off >>= 1) common += __shfl_xor(common, off, 32);
    if (common != 2) continue;  // manifold condition: exactly 2 shared neighbors

    --count;
    for (int i = 0; i < nw; ++i) {
      int c = i * 32 + lane;
      unsigned nwd = w0[i] | w1[i];
      if (c == (v0 >> 5)) nwd &= ~(1u << (v0 & 31));
      if (c == (v1 >> 5)) nwd &= ~(1u << (v1 & 31));
      r0[c] = nwd;
      r1[c] = 0u;
      unsigned s = nwd;  // scatter: each set bit is a unique neighbor row
      while (s) {
        int b = __ffs(s) - 1;
        s &= s - 1;
        int u = c * 32 + b;
        atomicOr (&U[(size_t)u * wpr + (v0 >> 5)],  1u << (v0 & 31));
        atomicAnd(&U[(size_t)u * wpr + (v1 >> 5)], ~(1u << (v1 & 31)));
      }
    }
    if (lane == 0) { mask[v1] = 0; collapsed[ptr] = 1; }
    __threadfence();  // make this collapse visible before next iteration's reads
  }
}

// ---------------------------------------------------------------------------
// 6) Replay: feature column f is owned by thread f -> per-thread program
//    order alone reproduces the sequential scan; no barriers needed.
// ---------------------------------------------------------------------------
__global__ __launch_bounds__(512) void replay_kernel(
    const unsigned long long* __restrict__ keys, const int* __restrict__ edges,
    const int* __restrict__ collapsed, float* feat, int E) {
  const int f = threadIdx.x;
  for (int ptr = 0; ptr < E; ++ptr) {
    if (ptr + 1 < E) __builtin_prefetch(&collapsed[ptr + 1], 0, 1);
    if (!collapsed[ptr]) continue;
    int eid = (int)(unsigned)(keys[ptr] & 0xFFFFFFFFull);
    int v0 = edges[eid], v1 = edges[E + eid];
    size_t i0 = (size_t)v0 * F_DIM + f;
    size_t i1 = (size_t)v1 * F_DIM + f;
    feat[i0] = 0.5f * (feat[i0] + feat[i1]);
  }
}

__global__ void copy_kernel(const float4* __restrict__ in, float4* __restrict__ out, size_t n4) {
  size_t i = (size_t)blockIdx.x * blockDim.x + threadIdx.x;
  if (i < n4) out[i] = in[i];
}

__global__ void mask_kernel(float4* out, const int* __restrict__ mask, size_t n4) {
  size_t i = (size_t)blockIdx.x * blockDim.x + threadIdx.x;
  if (i >= n4) return;
  int v = (int)(i / (F_DIM / 4));
  if (!mask[v]) out[i] = make_float4(0.f, 0.f, 0.f, 0.f);
}

// ---------------------------------------------------------------------------
extern "C" void kernel_launch(void* const* d_in, const int* in_sizes, int n_in,
                              void* d_out, int out_size, void* d_ws, size_t ws_size,
                              hipStream_t stream) {
  (void)n_in; (void)out_size; (void)ws_size;
  const float* image = (const float*)d_in[0];
  const int* edges   = (const int*)d_in[1];
  const float* vs    = (const float*)d_in[2];

  const int V = in_sizes[0] / F_DIM;      // 9216
  const int E = in_sizes[1] / 2;          // 27265
  int NSORT = 1;
  while (NSORT < E) NSORT <<= 1;          // 32768
  const int wpr = (((V + 31) / 32) + 31) & ~31;  // words/row, padded to 32 (288)
  const int nw = wpr / 32;                // 9 words per lane
  const size_t uwords = (size_t)V * wpr;  // 2.65M words = 10.6MB (L2-resident)

  char* ws = (char*)d_ws;
  size_t off = 0;
  auto alloc = [&](size_t bytes) -> void* {
    off = (off + 255) & ~(size_t)255;
    void* p = ws + off;
    off += bytes;
    return p;
  };
  float* sq                 = (float*)alloc((size_t)V * sizeof(float));
  unsigned long long* keys  = (unsigned long long*)alloc((size_t)NSORT * 8);
  unsigned* U               = (unsigned*)alloc(uwords * 4);
  int* mask                 = (int*)alloc((size_t)V * 4);
  int* bnd                  = (int*)alloc((size_t)V * 4);
  int* collapsed            = (int*)alloc((size_t)NSORT * 4);

  const size_t n4 = (size_t)V * (F_DIM / 4);

  copy_kernel<<<dim3((unsigned)((n4 + 255) / 256)), 256, 0, stream>>>(
      (const float4*)image, (float4*)d_out, n4);

  sq_kernel<<<dim3((unsigned)((V + 63) / 64)), 128, 0, stream>>>(image, sq, V);

  init_kernel<<<dim3((unsigned)((uwords + 255) / 256)), 256, 0, stream>>>(
      U, uwords, mask, bnd, collapsed, vs, V, NSORT);

  adj_kernel<<<dim3((unsigned)((E + 255) / 256)), 256, 0, stream>>>(edges, U, E, wpr);

  pack_kernel<<<dim3((unsigned)((NSORT + 255) / 256)), 256, 0, stream>>>(
      sq, edges, keys, E, NSORT);

  bitonic_kernel<<<1, 1024, (size_t)NSORT * 8, stream>>>(keys, NSORT);

  decimate_kernel<<<1, 32, 0, stream>>>(keys, edges, U, mask, bnd, collapsed,
                                        V, E, wpr, nw, V / 2);

  replay_kernel<<<1, F_DIM, 0, stream>>>(keys, edges, collapsed, (float*)d_out, E);

  mask_kernel<<<dim3((unsigned)((n4 + 255) / 256)), 256, 0, stream>>>(
      (float4*)d_out, mask, n4);
}